// GCNGraphBinClassifier_22299470201567
// MI455X (gfx1250) — compile-verified
//
#include <hip/hip_runtime.h>
#include <hip/hip_bf16.h>

#define HD 128          // hidden dim
#define NATOM_F 9
#define ATOM_VOCAB 119
#define NLAYERS 3

typedef float v2f __attribute__((ext_vector_type(2)));
typedef float v8f __attribute__((ext_vector_type(8)));

// ---------------------------------------------------------------- utilities
__global__ void zero_kernel(float* __restrict__ p, long n) {
  long i = (long)blockIdx.x * blockDim.x + threadIdx.x;
  long stride = (long)gridDim.x * blockDim.x;
  for (; i < n; i += stride) p[i] = 0.0f;
}

// ------------------------------------------------------------ atom encoder
// h[n,c] = sum_f atom_tables[f, x[n,f], c]
__global__ void atom_encoder_kernel(const int* __restrict__ x,
                                    const float* __restrict__ tables,
                                    float* __restrict__ h, int n_nodes) {
  int idx = blockIdx.x * blockDim.x + threadIdx.x;
  if (idx >= n_nodes * HD) return;
  int node = idx >> 7;
  int c = idx & (HD - 1);
  float s = 0.0f;
#pragma unroll
  for (int f = 0; f < NATOM_F; ++f) {
    int v = x[node * NATOM_F + f];
    s += tables[((long)f * ATOM_VOCAB + v) * HD + c];
  }
  h[idx] = s;
}

// ------------------------------------------------------------------ degree
__global__ void degree_kernel(const int* __restrict__ dst,
                              float* __restrict__ deg, int n_edges) {
  int e = blockIdx.x * blockDim.x + threadIdx.x;
  if (e < n_edges) atomicAdd(&deg[dst[e]], 1.0f);
}

__global__ void dinv_kernel(float* __restrict__ deg_dinv, int n_nodes) {
  int i = blockIdx.x * blockDim.x + threadIdx.x;
  if (i < n_nodes) deg_dinv[i] = rsqrtf(deg_dinv[i] + 1.0f);
}

// ------------------------------------------------- WMMA GEMM  hw = h @ W[l]
// block = 256 threads = 8 waves; blockIdx.x = 16-row tile (N % 16 == 0).
// Wave w computes the 16x16 output tile at columns [16w, 16w+16).
// f32 WMMA fragment layouts per CDNA5 ISA 7.12.2:
//   A 16x4 : lane(m=lane&15, half=lane>>4): v0 = A[m][2*half], v1 = A[m][2*half+1]
//   B 4x16 : v0 = B[2*half][n],  v1 = B[2*half+1][n]   (n = lane&15)
//   D 16x16: vr -> D[r + 8*half][n]
__global__ void gemm_wmma_kernel(const float* __restrict__ hin,
                                 const float* __restrict__ W,
                                 float* __restrict__ hw) {
  __shared__ float Atile[16][HD + 4];   // +4 pad -> conflict-free column reads
  int row0 = blockIdx.x * 16;
  for (int i = threadIdx.x; i < 16 * HD; i += blockDim.x) {
    int r = i >> 7, c = i & (HD - 1);
    Atile[r][c] = hin[(long)(row0 + r) * HD + c];
  }
  __syncthreads();

  int wave  = threadIdx.x >> 5;
  int lane  = threadIdx.x & 31;
  int m     = lane & 15;     // A row / B,D column within tile
  int khalf = lane >> 4;     // lane-half select
  int col0  = wave * 16;

  v8f acc = {0.f, 0.f, 0.f, 0.f, 0.f, 0.f, 0.f, 0.f};
  for (int kb = 0; kb < HD; kb += 4) {
    v2f a, b;
    a.x = Atile[m][kb + 2 * khalf];
    a.y = Atile[m][kb + 2 * khalf + 1];
    b.x = W[(kb + 2 * khalf) * HD + col0 + m];
    b.y = W[(kb + 2 * khalf + 1) * HD + col0 + m];
    acc = __builtin_amdgcn_wmma_f32_16x16x4_f32(false, a, false, b,
                                                (short)0, acc, false, false);
  }
#pragma unroll
  for (int r = 0; r < 8; ++r) {
    int row = row0 + r + 8 * khalf;
    hw[(long)row * HD + col0 + m] = acc[r];
  }
}

// ----------------------------------------------------- edge scatter (GCN)
// agg[dst,c] += hw[src,c] * dinv[src] * dinv[dst]
__global__ void scatter_kernel(const int* __restrict__ src,
                               const int* __restrict__ dst,
                               const float* __restrict__ dinv,
                               const float* __restrict__ hw,
                               float* __restrict__ agg, int n_edges) {
  int i = blockIdx.x * blockDim.x + threadIdx.x;
  if (i >= n_edges * HD) return;     // E*H = 102.4M < 2^31
  int e = i >> 7;
  int c = i & (HD - 1);
  int s = src[e], d = dst[e];
  float norm = dinv[s] * dinv[d];
  atomicAdd(&agg[(long)d * HD + c], hw[(long)s * HD + c] * norm);
}

// --------------------------------------------------------------- combine
// h = agg + hw * dinv^2 + bias ; optional ReLU
__global__ void combine_kernel(float* __restrict__ h,
                               const float* __restrict__ agg,
                               const float* __restrict__ hw,
                               const float* __restrict__ dinv,
                               const float* __restrict__ bias,
                               int n_nodes, int do_relu) {
  int idx = blockIdx.x * blockDim.x + threadIdx.x;
  if (idx >= n_nodes * HD) return;
  int node = idx >> 7;
  int c = idx & (HD - 1);
  float di = dinv[node];
  float v = agg[idx] + hw[idx] * (di * di) + bias[c];
  h[idx] = do_relu ? fmaxf(v, 0.0f) : v;
}

// ------------------------------------------------------------------ pool
__global__ void pool_accum_kernel(const float* __restrict__ h,
                                  const int* __restrict__ batch,
                                  float* __restrict__ sums,
                                  float* __restrict__ counts, int n_nodes) {
  int idx = blockIdx.x * blockDim.x + threadIdx.x;
  if (idx >= n_nodes * HD) return;
  int node = idx >> 7;
  int c = idx & (HD - 1);
  int g = batch[node];
  atomicAdd(&sums[(long)g * HD + c], h[idx]);
  if (c == 0) atomicAdd(&counts[g], 1.0f);
}

// out[g] = dot(sums[g]/max(counts[g],1), lm_w) + lm_b
__global__ void final_out_kernel(const float* __restrict__ sums,
                                 const float* __restrict__ counts,
                                 const float* __restrict__ lm_w,
                                 const float* __restrict__ lm_b,
                                 float* __restrict__ out) {
  __shared__ float red[HD];
  int g = blockIdx.x;
  int c = threadIdx.x;
  float cnt = fmaxf(counts[g], 1.0f);
  red[c] = (sums[(long)g * HD + c] / cnt) * lm_w[c];
  __syncthreads();
  for (int s = HD / 2; s > 0; s >>= 1) {
    if (c < s) red[c] += red[c + s];
    __syncthreads();
  }
  if (c == 0) out[g] = red[0] + lm_b[0];
}

// ------------------------------------------------------------------ driver
extern "C" void kernel_launch(void* const* d_in, const int* in_sizes, int n_in,
                              void* d_out, int out_size, void* d_ws, size_t ws_size,
                              hipStream_t stream) {
  const int*   x          = (const int*)  d_in[0];
  const int*   edge_index = (const int*)  d_in[1];
  // d_in[2] = edge_attr (dropped by reference), d_in[5] = bond_tables (dropped)
  const int*   batch      = (const int*)  d_in[3];
  const float* atom_tab   = (const float*)d_in[4];
  const float* conv_w     = (const float*)d_in[6];
  const float* conv_b     = (const float*)d_in[7];
  const float* lm_w       = (const float*)d_in[8];
  const float* lm_b       = (const float*)d_in[9];
  float*       out        = (float*)d_out;

  const int N = in_sizes[3];          // batch has N entries
  const int E = in_sizes[1] / 2;      // edge_index is [2, E]
  const int G = out_size;             // graphs in batch

  const int* src = edge_index;        // row 0
  const int* dst = edge_index + E;    // row 1

  // workspace layout (floats)
  float* ws     = (float*)d_ws;
  const long NH = (long)N * HD;
  float* h      = ws;                 // [N, H]
  float* hw     = h   + NH;           // [N, H]
  float* agg    = hw  + NH;           // [N, H]
  float* dinv   = agg + NH;           // [N]   (deg, then dinv in place)
  float* sums   = dinv + N;           // [G, H]
  float* counts = sums + (long)G * HD;// [G]

  const int T = 256;
  dim3 blkNH((int)((NH + T - 1) / T));
  dim3 blkEH((int)(((long)E * HD + T - 1) / T));

  // 1) atom encoder
  atom_encoder_kernel<<<blkNH, T, 0, stream>>>(x, atom_tab, h, N);

  // 2) degree -> dinv
  zero_kernel<<<512, T, 0, stream>>>(dinv, N);
  degree_kernel<<<(E + T - 1) / T, T, 0, stream>>>(dst, dinv, E);
  dinv_kernel<<<(N + T - 1) / T, T, 0, stream>>>(dinv, N);

  // 3) GCN layers
  for (int l = 0; l < NLAYERS; ++l) {
    gemm_wmma_kernel<<<N / 16, 256, 0, stream>>>(h, conv_w + (long)l * HD * HD, hw);
    zero_kernel<<<4096, T, 0, stream>>>(agg, NH);
    scatter_kernel<<<blkEH, T, 0, stream>>>(src, dst, dinv, hw, agg, E);
    combine_kernel<<<blkNH, T, 0, stream>>>(h, agg, hw, dinv,
                                            conv_b + (long)l * HD, N,
                                            (l < NLAYERS - 1) ? 1 : 0);
  }

  // 4) mean pool + linear head
  zero_kernel<<<64, T, 0, stream>>>(sums, (long)G * HD + G);  // sums then counts (contiguous)
  pool_accum_kernel<<<blkNH, T, 0, stream>>>(h, batch, sums, counts, N);
  final_out_kernel<<<G, HD, 0, stream>>>(sums, counts, lm_w, lm_b, out);
}